// CGCNN_52080773431371
// MI455X (gfx1250) — compile-verified
//
#include <hip/hip_runtime.h>
#include <math.h>

#define NN 100000
#define NE 1600000
#define NB 128
#define ND 91
#define NDP 96
#define ED 20
#define H 128
#define H2 256
#define NCV 3
#define EPSBN 1e-5f

typedef __attribute__((ext_vector_type(2))) float v2f;
typedef __attribute__((ext_vector_type(8))) float v8f;

// ---------------- generic fp32 GEMM via V_WMMA_F32_16X16X4_F32 -------------
// C(MxNc) = A(MxK) @ B(KxNc) + bias(Nc); M%16==0, K%4==0, Nc%64==0
// Each wave computes a 16x64 strip: one A fragment feeds 4 WMMAs per K-step.
__global__ __launch_bounds__(256) void gemm_wmma_f32(
    const float* __restrict__ A, const float* __restrict__ B,
    const float* __restrict__ bias, float* __restrict__ C,
    int M, int K, int Nc) {
  const int lane   = threadIdx.x & 31;
  const int wave   = blockIdx.x * (blockDim.x >> 5) + (threadIdx.x >> 5);
  const int tilesN = Nc >> 6;                       // strips of 64 columns
  const int tiles  = (M >> 4) * tilesN;
  if (wave >= tiles) return;
  const int row0 = (wave / tilesN) << 4;
  const int col0 = (wave % tilesN) << 6;
  const int r  = row0 + (lane & 15);                // A row for this lane
  const int c  = col0 + (lane & 15);                // first B/C column
  const int kh = (lane >> 4) << 1;                  // K sub-offset per half
  v8f acc0 = {}, acc1 = {}, acc2 = {}, acc3 = {};
  for (int k = 0; k < K; k += 4) {
    const int kk = k + kh;
    const float2 av = *(const float2*)(A + (size_t)r * K + kk);
    v2f a;
    a.x = av.x;
    a.y = av.y;
    const float* b0 = B + (size_t)kk * Nc + c;
    const float* b1 = B + (size_t)(kk + 1) * Nc + c;
    v2f w0, w1, w2, w3;
    w0.x = b0[0];  w0.y = b1[0];
    w1.x = b0[16]; w1.y = b1[16];
    w2.x = b0[32]; w2.y = b1[32];
    w3.x = b0[48]; w3.y = b1[48];
    acc0 = __builtin_amdgcn_wmma_f32_16x16x4_f32(false, a, false, w0, (short)0, acc0, false, false);
    acc1 = __builtin_amdgcn_wmma_f32_16x16x4_f32(false, a, false, w1, (short)0, acc1, false, false);
    acc2 = __builtin_amdgcn_wmma_f32_16x16x4_f32(false, a, false, w2, (short)0, acc2, false, false);
    acc3 = __builtin_amdgcn_wmma_f32_16x16x4_f32(false, a, false, w3, (short)0, acc3, false, false);
  }
  float bv0 = 0.0f, bv1 = 0.0f, bv2 = 0.0f, bv3 = 0.0f;
  if (bias) {
    bv0 = bias[c];
    bv1 = bias[c + 16];
    bv2 = bias[c + 32];
    bv3 = bias[c + 48];
  }
  const int rbase = row0 + ((lane >> 4) << 3);
#pragma unroll
  for (int j = 0; j < 8; ++j) {
    float* crow = C + (size_t)(rbase + j) * Nc + c;
    crow[0]  = acc0[j] + bv0;
    crow[16] = acc1[j] + bv1;
    crow[32] = acc2[j] + bv2;
    crow[48] = acc3[j] + bv3;
  }
}

// ---------------- helpers ---------------------------------------------------
__global__ void zero_f32(float* __restrict__ p, size_t n) {
  size_t i = (size_t)blockIdx.x * blockDim.x + threadIdx.x;
  size_t st = (size_t)gridDim.x * blockDim.x;
  for (; i < n; i += st) p[i] = 0.0f;
}

__global__ void pad_nodefeats(const float* __restrict__ nf, float* __restrict__ Ap) {
  size_t i = (size_t)blockIdx.x * blockDim.x + threadIdx.x;
  size_t st = (size_t)gridDim.x * blockDim.x;
  const size_t tot = (size_t)NN * NDP;
  for (; i < tot; i += st) {
    int c = (int)(i % NDP);
    size_t r = i / NDP;
    Ap[i] = (c < ND) ? nf[r * ND + c] : 0.0f;
  }
}

__global__ void pad_embedw(const float* __restrict__ w, float* __restrict__ Wp) {
  int i = blockIdx.x * blockDim.x + threadIdx.x;   // NDP*H elements
  if (i < NDP * H) Wp[i] = (i < ND * H) ? w[i] : 0.0f;
}

__device__ __forceinline__ float sig_(float x) { return 1.0f / (1.0f + __expf(-x)); }
__device__ __forceinline__ float sp_(float x)  { return fmaxf(x, 0.0f) + log1pf(__expf(-fabsf(x))); }

// ---------------- fused edge kernel ----------------------------------------
// z = hs[src] + hd[dst] + ef@we ; msg = sig(z[:H])*softplus(z[H:]) ; atomic agg
__global__ __launch_bounds__(256) void edge_kernel(
    const int* __restrict__ src, const int* __restrict__ dst,
    const float* __restrict__ ef, const float* __restrict__ we,
    const float* __restrict__ hs, const float* __restrict__ hd,
    float* __restrict__ agg) {
  __shared__ __align__(16) float we_l[ED * H2];   // 20 KB
  for (int i = threadIdx.x; i < ED * H2; i += blockDim.x) we_l[i] = we[i];
  __syncthreads();
  const int lane = threadIdx.x & 31;
  const int wpb  = blockDim.x >> 5;
  const long w0  = (long)blockIdx.x * wpb + (threadIdx.x >> 5);
  const long nw  = (long)gridDim.x * wpb;
  const int c = lane << 2;                        // 4 gate cols per lane
  for (long e = w0; e < NE; e += nw) {
    const int s = src[e], d = dst[e];
    float efv = (lane < ED) ? ef[(size_t)e * ED + lane] : 0.0f;
    float4 zg = *(const float4*)(hs + (size_t)s * H2 + c);
    float4 zf = *(const float4*)(hs + (size_t)s * H2 + H + c);
    const float4 tg = *(const float4*)(hd + (size_t)d * H2 + c);
    const float4 tf = *(const float4*)(hd + (size_t)d * H2 + H + c);
    zg.x += tg.x; zg.y += tg.y; zg.z += tg.z; zg.w += tg.w;
    zf.x += tf.x; zf.y += tf.y; zf.z += tf.z; zf.w += tf.w;
#pragma unroll
    for (int t = 0; t < ED; ++t) {
      const float ev = __shfl(efv, t, 32);
      const float4 wg = *(const float4*)(we_l + t * H2 + c);
      const float4 wf = *(const float4*)(we_l + t * H2 + H + c);
      zg.x += ev * wg.x; zg.y += ev * wg.y; zg.z += ev * wg.z; zg.w += ev * wg.w;
      zf.x += ev * wf.x; zf.y += ev * wf.y; zf.z += ev * wf.z; zf.w += ev * wf.w;
    }
    float* a = agg + (size_t)d * H + c;
    atomicAdd(a + 0, sig_(zg.x) * sp_(zf.x));
    atomicAdd(a + 1, sig_(zg.y) * sp_(zf.y));
    atomicAdd(a + 2, sig_(zg.z) * sp_(zf.z));
    atomicAdd(a + 3, sig_(zg.w) * sp_(zf.w));
  }
}

// ---------------- batchnorm (training-mode, biased var) ---------------------
__global__ void colstats(const float* __restrict__ x, int n,
                         float* __restrict__ sum, float* __restrict__ sumsq) {
  const int c = threadIdx.x;                      // H threads
  float s = 0.0f, s2 = 0.0f;
  for (int r = blockIdx.x; r < n; r += gridDim.x) {
    const float v = x[(size_t)r * H + c];
    s += v; s2 += v * v;
  }
  atomicAdd(&sum[c], s);
  atomicAdd(&sumsq[c], s2);
}

__global__ void bn_finalize(const float* __restrict__ sum, const float* __restrict__ sumsq,
                            const float* __restrict__ gamma, const float* __restrict__ beta,
                            float n, float* __restrict__ scale, float* __restrict__ shift) {
  const int c = threadIdx.x;
  const float m  = sum[c] / n;
  const float v  = sumsq[c] / n - m * m;
  const float sc = gamma[c] * rsqrtf(v + EPSBN);
  scale[c] = sc;
  shift[c] = beta[c] - m * sc;
}

__global__ void bn_apply_relu(float* __restrict__ h, const float* __restrict__ agg,
                              const float* __restrict__ scale, const float* __restrict__ shift) {
  size_t i = (size_t)blockIdx.x * blockDim.x + threadIdx.x;
  const size_t tot = (size_t)NN * H;
  const size_t st = (size_t)gridDim.x * blockDim.x;
  for (; i < tot; i += st) {
    const int c = (int)(i & (H - 1));
    h[i] = fmaxf(h[i] + agg[i] * scale[c] + shift[c], 0.0f);
  }
}

// ---------------- pooling & head -------------------------------------------
__global__ void pool_accum(const float* __restrict__ h, const int* __restrict__ batch,
                           float* __restrict__ gsum, float* __restrict__ gcnt) {
  size_t i = (size_t)blockIdx.x * blockDim.x + threadIdx.x;
  const size_t tot = (size_t)NN * H;
  const size_t st = (size_t)gridDim.x * blockDim.x;
  for (; i < tot; i += st) {
    const size_t n = i >> 7;
    const int c = (int)(i & (H - 1));
    const int b = batch[n];
    atomicAdd(&gsum[(size_t)b * H + c], h[i]);
    if (c == 0) atomicAdd(&gcnt[b], 1.0f);
  }
}

__global__ void pool_div(const float* __restrict__ gsum, const float* __restrict__ gcnt,
                         float* __restrict__ g) {
  const int i = blockIdx.x * blockDim.x + threadIdx.x;
  if (i < NB * H) g[i] = gsum[i] / fmaxf(gcnt[i >> 7], 1.0f);
}

__global__ void fc_out(const float* __restrict__ y, const float* __restrict__ scale,
                       const float* __restrict__ shift, const float* __restrict__ ow,
                       const float* __restrict__ ob, float* __restrict__ out) {
  const int b = threadIdx.x;                      // NB threads
  float acc = 0.0f;
  for (int c = 0; c < H; ++c) {
    const float x = fmaxf(y[(size_t)b * H + c] * scale[c] + shift[c], 0.0f);
    acc += x * ow[c];
  }
  out[b] = acc + ob[0];
}

// ---------------- orchestration --------------------------------------------
extern "C" void kernel_launch(void* const* d_in, const int* in_sizes, int n_in,
                              void* d_out, int out_size, void* d_ws, size_t ws_size,
                              hipStream_t stream) {
  const float* node_feats = (const float*)d_in[0];
  const int*   eidx       = (const int*)d_in[1];
  const float* edge_feats = (const float*)d_in[2];
  const int*   batch      = (const int*)d_in[3];
  const float* embed_w    = (const float*)d_in[4];
  const float* embed_b    = (const float*)d_in[5];
  const float* conv_wsrc  = (const float*)d_in[6];
  const float* conv_wdst  = (const float*)d_in[7];
  const float* conv_we    = (const float*)d_in[8];
  const float* conv_b     = (const float*)d_in[9];
  const float* bn_gamma   = (const float*)d_in[10];
  const float* bn_beta    = (const float*)d_in[11];
  const float* fc1_w      = (const float*)d_in[12];
  const float* fc1_b      = (const float*)d_in[13];
  const float* fc_bn_g    = (const float*)d_in[14];
  const float* fc_bn_b    = (const float*)d_in[15];
  const float* out_w      = (const float*)d_in[16];
  const float* out_b      = (const float*)d_in[17];
  const int* src = eidx;
  const int* dst = eidx + NE;

  // workspace layout (floats); total ~308 MB
  float* ws   = (float*)d_ws;
  float* h    = ws;                          // NN*H
  float* hs   = h   + (size_t)NN * H;        // NN*H2
  float* hd   = hs  + (size_t)NN * H2;       // NN*H2
  float* agg  = hd  + (size_t)NN * H2;       // NN*H   (reused as padded A: NN*NDP <= NN*H)
  float* sm   = agg + (size_t)NN * H;        // small tail
  float* colsum = sm;                        // H
  float* colsq  = sm + H;                    // H
  float* scale  = sm + 2 * H;                // H
  float* shift  = sm + 3 * H;                // H
  float* gsum   = sm + 4 * H;                // NB*H
  float* gcnt   = gsum + (size_t)NB * H;     // NB
  float* g      = gcnt + NB;                 // NB*H
  float* y      = g + (size_t)NB * H;        // NB*H
  float* Wpad   = y + (size_t)NB * H;        // NDP*H
  float* Apad   = agg;                       // alias (dead until conv layers)

  // 1) embed: h = pad(node_feats) @ pad(embed_w) + embed_b
  pad_nodefeats<<<4096, 256, 0, stream>>>(node_feats, Apad);
  pad_embedw<<<(NDP * H + 255) / 256, 256, 0, stream>>>(embed_w, Wpad);
  {
    const int tiles = (NN / 16) * (H / 64);
    gemm_wmma_f32<<<(tiles + 7) / 8, 256, 0, stream>>>(Apad, Wpad, embed_b, h, NN, NDP, H);
  }

  // 2) conv layers
  for (int i = 0; i < NCV; ++i) {
    const float* wsrc = conv_wsrc + (size_t)i * H * H2;
    const float* wdst = conv_wdst + (size_t)i * H * H2;
    const float* we   = conv_we   + (size_t)i * ED * H2;
    const float* cb   = conv_b    + (size_t)i * H2;
    const int tiles = (NN / 16) * (H2 / 64);
    gemm_wmma_f32<<<(tiles + 7) / 8, 256, 0, stream>>>(h, wsrc, cb,      hs, NN, H, H2);
    gemm_wmma_f32<<<(tiles + 7) / 8, 256, 0, stream>>>(h, wdst, nullptr, hd, NN, H, H2);
    zero_f32<<<4096, 256, 0, stream>>>(agg, (size_t)NN * H);
    edge_kernel<<<2048, 256, 0, stream>>>(src, dst, edge_feats, we, hs, hd, agg);
    zero_f32<<<1, 256, 0, stream>>>(colsum, 2 * H);
    colstats<<<512, H, 0, stream>>>(agg, NN, colsum, colsq);
    bn_finalize<<<1, H, 0, stream>>>(colsum, colsq, bn_gamma + (size_t)i * H,
                                     bn_beta + (size_t)i * H, (float)NN, scale, shift);
    bn_apply_relu<<<4096, 256, 0, stream>>>(h, agg, scale, shift);
  }

  // 3) global mean pool
  zero_f32<<<65, 256, 0, stream>>>(gsum, (size_t)NB * H + NB);   // gsum + gcnt contiguous
  pool_accum<<<4096, 256, 0, stream>>>(h, batch, gsum, gcnt);
  pool_div<<<(NB * H + 255) / 256, 256, 0, stream>>>(gsum, gcnt, g);

  // 4) head: y = g@fc1_w + fc1_b ; bn ; relu ; @out_w + out_b
  {
    const int tiles = (NB / 16) * (H / 64);
    gemm_wmma_f32<<<(tiles + 7) / 8, 256, 0, stream>>>(g, fc1_w, fc1_b, y, NB, H, H);
  }
  zero_f32<<<1, 256, 0, stream>>>(colsum, 2 * H);
  colstats<<<64, H, 0, stream>>>(y, NB, colsum, colsq);
  bn_finalize<<<1, H, 0, stream>>>(colsum, colsq, fc_bn_g, fc_bn_b, (float)NB, scale, shift);
  fc_out<<<1, H, 0, stream>>>(y, scale, shift, out_w, out_b, (float*)d_out);
}